// MMCE_4587025072616
// MI455X (gfx1250) — compile-verified
//
#include <hip/hip_runtime.h>
#include <hip/hip_bf16.h>
#include <stdint.h>

// MMCE: phase 1 bandwidth-bound row softmax stats (32MB @ 23.3TB/s ~ 1.4us),
// phase 2 TRANS-bound 67M v_exp_f32 pairwise sum with LDS-staged tiles.
// CDNA5 paths: async global->LDS staging (ASYNCcnt) + WMMA f32 16x16x4
// matrix-unit wave reduction.

#define NROWS 8192
#define NCOLS 1000
#define JT    1024           // j-tile staged in LDS per block (8 KB total)
#define KBW_INV 2.5f         // 1 / 0.4

typedef __attribute__((ext_vector_type(2))) float v2f;
typedef __attribute__((ext_vector_type(8))) float v8f;
typedef __attribute__((__vector_size__(16))) int v4i;      // builtin's 128-bit unit
typedef __attribute__((address_space(3)))    v4i Lv4i;     // LDS-side int4

#if defined(__has_builtin)
# if __has_builtin(__builtin_amdgcn_global_load_async_to_lds_b128)
#  define HAVE_ASYNC_B128 1
# endif
# if __has_builtin(__builtin_amdgcn_s_wait_asynccnt)
#  define HAVE_WAIT_ASYNC 1
# endif
# if __has_builtin(__builtin_amdgcn_wmma_f32_16x16x4_f32)
#  define HAVE_WMMA_F32X4 1
# endif
#endif

// ---- wave32 full reduction through the matrix unit ----------------------
// A[m][0]=acc(lane m), A[m][2]=acc(lane m+16), other K slots 0; B = ones.
// D[m][n] = acc[m] + acc[m+16]. Lane L<16 holds rows 0..7 of column L,
// lane L>=16 holds rows 8..15; summing 8 D regs + shfl_xor(16) = wave sum.
__device__ __forceinline__ float wave_sum32(float acc) {
#if defined(HAVE_WMMA_F32X4)
    v2f A; A.x = acc;  A.y = 0.0f;
    v2f B; B.x = 1.0f; B.y = 1.0f;
    v8f Cz = {};
    v8f D = __builtin_amdgcn_wmma_f32_16x16x4_f32(
        false, A, false, B, (short)0, Cz, false, false);
    float r = ((D[0] + D[1]) + (D[2] + D[3])) + ((D[4] + D[5]) + (D[6] + D[7]));
    r += __shfl_xor(r, 16, 32);
    return r;
#else
    for (int off = 16; off > 0; off >>= 1) acc += __shfl_xor(acc, off, 32);
    return acc;
#endif
}

// ---- Kernel 1: per-row softmax max-prob + correctness mask ---------------
// One wave per row, 8 waves/block. Two passes (row re-read hits L2).
__global__ void k_row_stats(const float* __restrict__ logits,
                            const int*   __restrict__ target,
                            float* __restrict__ p_out,
                            float* __restrict__ v_out) {
    const int wave = threadIdx.x >> 5;
    const int lane = threadIdx.x & 31;
    const int row  = blockIdx.x * 8 + wave;
    const float* x = logits + (size_t)row * NCOLS;

    float m = -INFINITY; int mi = 0x7fffffff;
    for (int c = lane; c < NCOLS; c += 32) {
        float val = x[c];
        if (val > m) { m = val; mi = c; }
    }
    // wave reduce (max, first-index tie-break to match jnp.argmax)
    for (int off = 16; off > 0; off >>= 1) {
        float om = __shfl_xor(m, off, 32);
        int   oi = __shfl_xor(mi, off, 32);
        if (om > m || (om == m && oi < mi)) { m = om; mi = oi; }
    }
    float s = 0.0f;
    for (int c = lane; c < NCOLS; c += 32) s += __expf(x[c] - m);
    for (int off = 16; off > 0; off >>= 1) s += __shfl_xor(s, off, 32);

    if (lane == 0) {
        float p = 1.0f / s;                       // exp(m-m)/sum = max softmax prob
        float cm = (mi == target[row]) ? 1.0f : 0.0f;
        p_out[row] = p;
        v_out[row] = cm - p;
    }
}

// ---- Kernel 2: tiled pairwise Laplacian-kernel sum -----------------------
// grid = (32 i-blocks of 256 rows, 8 j-chunks of 1024). Each thread owns one
// i, loops the LDS-staged j tile: acc += v_j * exp(-|p_i-p_j|*2.5).
__global__ void k_pairwise(const float* __restrict__ p,
                           const float* __restrict__ v,
                           float* __restrict__ partial) {
    __shared__ float s_p[JT];
    __shared__ float s_v[JT];
    __shared__ float s_wave[8];

    const int tid   = threadIdx.x;
    const int i     = blockIdx.x * 256 + tid;
    const int jbase = blockIdx.y * JT;

    const float pi = p[i];
    const float vi = v[i];

#if defined(HAVE_ASYNC_B128) && defined(HAVE_WAIT_ASYNC)
    {   // CDNA5 async global->LDS: 256 lanes x 16B = 4KB per array
        const float* gp = p + jbase + tid * 4;
        const float* gv = v + jbase + tid * 4;
        Lv4i* lp = (Lv4i*)(uint32_t)(uintptr_t)(s_p + tid * 4);
        Lv4i* lv = (Lv4i*)(uint32_t)(uintptr_t)(s_v + tid * 4);
        __builtin_amdgcn_global_load_async_to_lds_b128(
            (v4i*)(uintptr_t)gp, lp, 0, 0);
        __builtin_amdgcn_global_load_async_to_lds_b128(
            (v4i*)(uintptr_t)gv, lv, 0, 0);
        __builtin_amdgcn_s_wait_asynccnt(0);
    }
#else
    for (int k = tid * 4; k < JT; k += 256 * 4) {
        *(float4*)(s_p + k) = *(const float4*)(p + jbase + k);
        *(float4*)(s_v + k) = *(const float4*)(v + jbase + k);
    }
#endif
    __syncthreads();

    float acc = 0.0f;
#pragma unroll 8
    for (int k = 0; k < JT; ++k) {
        float d = pi - s_p[k];
        acc += s_v[k] * __expf(-fabsf(d) * KBW_INV);
    }
    acc *= vi;

    // wave sum via matrix unit, then fixed-order block sum
    float w = wave_sum32(acc);
    if ((tid & 31) == 0) s_wave[tid >> 5] = w;
    __syncthreads();
    if (tid == 0) {
        float t = 0.0f;
#pragma unroll
        for (int q = 0; q < 8; ++q) t += s_wave[q];
        partial[blockIdx.x * 8 + blockIdx.y] = t;
    }
}

// ---- Kernel 3: deterministic final reduce of 256 partials ----------------
__global__ void k_final(const float* __restrict__ partial,
                        float* __restrict__ out) {
    __shared__ float s[8];
    const int tid = threadIdx.x;
    float x = partial[tid];
    float w = wave_sum32(x);
    if ((tid & 31) == 0) s[tid >> 5] = w;
    __syncthreads();
    if (tid == 0) {
        float t = 0.0f;
#pragma unroll
        for (int q = 0; q < 8; ++q) t += s[q];
        out[0] = t / ((float)NROWS * (float)NROWS);
    }
}

extern "C" void kernel_launch(void* const* d_in, const int* in_sizes, int n_in,
                              void* d_out, int out_size, void* d_ws, size_t ws_size,
                              hipStream_t stream) {
    const float* logits = (const float*)d_in[0];
    const int*   target = (const int*)d_in[1];

    float* p_buf   = (float*)d_ws;           // [8192]
    float* v_buf   = p_buf + NROWS;          // [8192]
    float* partial = v_buf + NROWS;          // [256]
    float* out     = (float*)d_out;

    k_row_stats<<<NROWS / 8, 256, 0, stream>>>(logits, target, p_buf, v_buf);

    dim3 g2(NROWS / 256, NROWS / JT);        // (32, 8)
    k_pairwise<<<g2, 256, 0, stream>>>(p_buf, v_buf, partial);

    k_final<<<1, 256, 0, stream>>>(partial, out);
}